// MPNN_77592879169922
// MI455X (gfx1250) — compile-verified
//
#include <hip/hip_runtime.h>
#include <math.h>

// MPNN on MI455X (gfx1250, wave32).
// Sizes: N=768 nodes, IN=64, HID=128, EF=16, T=3.
// - All GEMMs via V_WMMA_F32_16X16X4_F32 (exact f32 semantics).
// - The 302MB e_msg tensor is never materialized: recomputed as a K=16 WMMA
//   chain fused with the max-reduction while streaming edges_mat (37.7MB,
//   L2-resident after first pass -> compute-bound, not HBM-bound).
// - Edge tiles are staged into LDS by the Tensor Data Mover (TDM), with
//   TDM-side padding producing a bank-conflict-free stride-18 layout,
//   double-buffered against the WMMA chain (TENSORcnt + block barrier).

#define N_NODES 768
#define IN_F    64
#define HID     128
#define EF      16
#define T_STEPS 3

typedef __attribute__((ext_vector_type(2))) float        v2f;
typedef __attribute__((ext_vector_type(8))) float        v8f;
typedef __attribute__((ext_vector_type(4))) unsigned int u32x4;
typedef __attribute__((ext_vector_type(8))) int          i32x8;
typedef __attribute__((ext_vector_type(4))) int          i32x4;

// ---------------------------------------------------------------------------
// TDM: DMA a 16(rows) x 16(f32) tile, row stride 16 elements, global -> LDS.
// LDS padding: after every 16 DWORDs insert 2 DWORDs => LDS row stride 18
// (pad_interval code 3 = 8 qwords = 16 dwords; pad_amount code 1 = 2 dwords).
// D# layout per CDNA5 ISA.
// ---------------------------------------------------------------------------
__device__ inline void tdm_load_tile16x16(const float* gsrc, unsigned lds_off) {
    unsigned long long ga = (unsigned long long)(uintptr_t)gsrc;

    u32x4 g0;
    g0[0] = 1u;                                        // count=1, user desc
    g0[1] = lds_off;                                   // lds_addr (bytes)
    g0[2] = (unsigned)ga;                              // global_addr[31:0]
    g0[3] = (unsigned)((ga >> 32) & 0x01FFFFFFu)       // global_addr[56:32]
            | (2u << 30);                              // type = 2 ("image")

    i32x8 g1;
    g1[0] = (int)((2u << 16)     // data_size = 2 -> 4 bytes
                | (1u << 20)     // pad_enable
                | (3u << 22)     // pad_interval code 3 -> every 16 dwords
                | (1u << 25));   // pad_amount code 1 -> 2 dwords
    g1[1] = (int)(16u << 16);                          // tensor_dim0 = 16 (lo16)
    g1[2] = 0;                                         // dim0 hi / dim1 lo16 (589824 -> lo16=0)
    g1[3] = (int)(0x9u | (16u << 16));                 // tensor_dim1 hi16 = 9, tile_dim0 = 16
    g1[4] = 16;                                        // tile_dim1 = 16, tile_dim2 = 0
    g1[5] = 16;                                        // tensor_dim0_stride = 16 elements
    g1[6] = 0;
    g1[7] = 0;

    i32x4 zero4 = {0, 0, 0, 0};                        // <=2D: groups 2/3 unused
#if __clang_major__ >= 23
    i32x8 zero8 = {0, 0, 0, 0, 0, 0, 0, 0};
    __builtin_amdgcn_tensor_load_to_lds(g0, g1, zero4, zero4, zero8, 0);
#else
    __builtin_amdgcn_tensor_load_to_lds(g0, g1, zero4, zero4, 0);
#endif
}

// ---------------------------------------------------------------------------
// Accumulate a 16x16 f32 tile: acc += X[row, 0:kcnt] @ W[wrow0: , col tile].
// Lane layout per CDNA5 ISA:
//   A 16x4 frag : lane L -> m=L&15, VGPR0=K(kb), VGPR1=K(kb+1), kb=(L>>4)*2
//   B 4x16 frag : lane L -> n=L&15, VGPR0=K(kb), VGPR1=K(kb+1)
//   C/D 16x16   : lane L, VGPR v -> (m = v + 8*(L>>4), n = L&15)
// ---------------------------------------------------------------------------
__device__ inline v8f wmma_accum_f32(v8f acc,
                                     const float* __restrict__ X, int ldx, int kcnt,
                                     const float* __restrict__ W, int ldw, int wrow0,
                                     int rowm, int kb, int n) {
    for (int k0 = 0; k0 < kcnt; k0 += 4) {
        const float* ap = X + (size_t)rowm * ldx + k0 + kb;
        v2f A; A.x = ap[0]; A.y = ap[1];
        const float* bp = W + (size_t)(wrow0 + k0 + kb) * ldw + n;
        v2f B; B.x = bp[0]; B.y = bp[ldw];
        acc = __builtin_amdgcn_wmma_f32_16x16x4_f32(false, A, false, B,
                                                    (short)0, acc, false, false);
    }
    return acc;
}

// D[row,col] = [X1 | X2] @ W + bias.  grid = (M/16, Nout/16), block = 32 (1 wave).
__global__ void gemm16_wmma(const float* __restrict__ X1, int ldx1, int k1,
                            const float* __restrict__ X2, int ldx2, int k2,
                            const float* __restrict__ W,  int ldw,
                            const float* __restrict__ bias,
                            float* __restrict__ D, int ldd) {
    int lane = threadIdx.x & 31;
    int m    = lane & 15;
    int half = lane >> 4;
    int kb   = half * 2;
    int n    = m;
    int row0 = blockIdx.x * 16;
    int col0 = blockIdx.y * 16;

    v8f acc = {};
    acc = wmma_accum_f32(acc, X1, ldx1, k1, W + col0, ldw, 0,  row0 + m, kb, n);
    if (X2)
        acc = wmma_accum_f32(acc, X2, ldx2, k2, W + col0, ldw, k1, row0 + m, kb, n);

    float bv = bias ? bias[col0 + n] : 0.0f;
#pragma unroll
    for (int v = 0; v < 8; ++v) {
        int row = row0 + v + 8 * half;
        D[(size_t)row * ldd + col0 + n] = acc[v] + bv;
    }
}

// ---------------------------------------------------------------------------
// u[j,h] = bprime[j,h] + max_i( a[i,h] + (edges[i,j,:] @ Wm_e)[h] )
// grid = 48 (j-tiles of 16), block = 256 = 8 waves (h-tiles of 16 -> 128 h).
// Edge tile staged by TDM into double-buffered LDS (stride 18 via TDM pad);
// a[i,:] row staged by waves 0-3. One barrier per iteration.
// ---------------------------------------------------------------------------
__global__ void msgmax_kernel(const float* __restrict__ edges,   // (768,768,16)
                              const float* __restrict__ Wme,     // (16,128) = Wm+256*128
                              const float* __restrict__ a,       // (768,128)
                              const float* __restrict__ bprime,  // (768,128) incl. bm
                              float* __restrict__ u) {           // (768,128)
    __shared__ float Es[2][16 * 18];
    __shared__ float As[2][128];

    int tid  = threadIdx.x;
    int lane = tid & 31;
    int w    = tid >> 5;               // wave id = h-tile
    int m    = lane & 15;
    int half = lane >> 4;
    int kb   = half * 2;
    int n    = m;
    int j0   = blockIdx.x * 16;
    int col  = w * 16 + n;             // global h index for this lane

    unsigned ldsEs0 = (unsigned)(uintptr_t)&Es[0][0];  // LDS byte offsets
    unsigned ldsEs1 = (unsigned)(uintptr_t)&Es[1][0];

    // B fragments (Wm_e[:, col tile]) are loop-invariant: preload 4 K-chunks.
    v2f Bf[4];
#pragma unroll
    for (int f = 0; f < 4; ++f) {
        Bf[f].x = Wme[(size_t)(4 * f + kb) * HID + col];
        Bf[f].y = Wme[(size_t)(4 * f + kb + 1) * HID + col];
    }

    // Stage i = 0.
    if (w == 0) tdm_load_tile16x16(edges + (size_t)j0 * EF, ldsEs0);
    if (tid < HID) As[0][tid] = a[tid];
    __builtin_amdgcn_s_wait_tensorcnt(0);
    __syncthreads();

    float mv[8];
#pragma unroll
    for (int v = 0; v < 8; ++v) mv[v] = -3.402823e38f;

    for (int i = 0; i < N_NODES; ++i) {
        int cur = i & 1;
        if (i + 1 < N_NODES) {
            if (w == 0)
                tdm_load_tile16x16(edges + ((size_t)(i + 1) * N_NODES + j0) * EF,
                                   cur ? ldsEs0 : ldsEs1);
            if (tid < HID) As[cur ^ 1][tid] = a[(size_t)(i + 1) * HID + tid];
        }

        // 16x16 e_msg tile for this (i, j-tile, h-tile): K=16 = 4 chained WMMAs.
        v8f d = {};
        const float* er = &Es[cur][m * 18 + kb];
#pragma unroll
        for (int f = 0; f < 4; ++f) {
            v2f A; A.x = er[4 * f]; A.y = er[4 * f + 1];
            d = __builtin_amdgcn_wmma_f32_16x16x4_f32(false, A, false, Bf[f],
                                                      (short)0, d, false, false);
        }
        float s = As[cur][col];        // a[i, col] broadcast within half-wave
#pragma unroll
        for (int v = 0; v < 8; ++v) mv[v] = fmaxf(mv[v], d[v] + s);

        __builtin_amdgcn_s_wait_tensorcnt(0);   // wave 0 drains its DMA
        __syncthreads();                        // publish next buffer to all waves
    }

#pragma unroll
    for (int v = 0; v < 8; ++v) {
        int row = j0 + v + 8 * half;
        u[(size_t)row * HID + col] = mv[v] + bprime[(size_t)row * HID + col];
    }
}

// ---------------------------------------------------------------------------
// Small helper kernels
// ---------------------------------------------------------------------------
__global__ void setup_kernel(const float* __restrict__ state0,
                             float* __restrict__ H,
                             float* __restrict__ preds,
                             float* __restrict__ pstop) {
    int idx = blockIdx.x * 256 + threadIdx.x;             // 0 .. 98303
    if (idx < N_NODES * HID) H[idx] = 0.0f;
    if (idx < N_NODES * IN_F) {
        int nd = idx >> 6, c = idx & 63;
        preds[(size_t)nd * ((T_STEPS + 1) * IN_F) + c] = state0[idx];
    }
    if (idx < N_NODES * 2)
        pstop[(size_t)(idx >> 1) * ((T_STEPS + 1) * 2) + (idx & 1)] = 0.0f;
}

__global__ void mean_kernel(const float* __restrict__ H, float* __restrict__ meanv) {
    int h = threadIdx.x;                                   // 128 threads
    float s = 0.0f;
    for (int nd = 0; nd < N_NODES; ++nd) s += H[(size_t)nd * HID + h];
    meanv[h] = s * (1.0f / (float)N_NODES);
}

__global__ void stop_kernel(const float* __restrict__ H,
                            const float* __restrict__ meanv,
                            const float* __restrict__ Wt,   // (256,2)
                            const float* __restrict__ bt,   // (2)
                            float* __restrict__ pstop) {    // base already offset to (t+1)
    int nd = blockIdx.x * 256 + threadIdx.x;
    if (nd >= N_NODES) return;
    float s0 = bt[0], s1 = bt[1];
    for (int k = 0; k < HID; ++k) {
        float hv = H[(size_t)nd * HID + k];
        s0 += hv * Wt[k * 2 + 0];
        s1 += hv * Wt[k * 2 + 1];
    }
    for (int k = 0; k < HID; ++k) {
        float mvv = meanv[k];
        s0 += mvv * Wt[(HID + k) * 2 + 0];
        s1 += mvv * Wt[(HID + k) * 2 + 1];
    }
    s0 = 1.0f / (1.0f + __expf(-s0));
    s1 = 1.0f / (1.0f + __expf(-s1));
    pstop[(size_t)nd * ((T_STEPS + 1) * 2) + 0] = s0;
    pstop[(size_t)nd * ((T_STEPS + 1) * 2) + 1] = s1;
}

// ---------------------------------------------------------------------------
extern "C" void kernel_launch(void* const* d_in, const int* in_sizes, int n_in,
                              void* d_out, int out_size, void* d_ws, size_t ws_size,
                              hipStream_t stream) {
    const float* states = (const float*)d_in[0];   // (3,768,64)
    const float* edges  = (const float*)d_in[1];   // (768,768,16)
    const float* We     = (const float*)d_in[2];   // (192,128)
    const float* be     = (const float*)d_in[3];   // (128)
    const float* Wm     = (const float*)d_in[4];   // (272,128)
    const float* bm     = (const float*)d_in[5];   // (128)
    const float* Wu     = (const float*)d_in[6];   // (256,128)
    const float* bu     = (const float*)d_in[7];   // (128)
    const float* Wd     = (const float*)d_in[8];   // (256,64)
    const float* bd     = (const float*)d_in[9];   // (64)
    const float* Wt     = (const float*)d_in[10];  // (256,2)
    const float* bt     = (const float*)d_in[11];  // (2)

    float* preds = (float*)d_out;                       // (768, 4, 64)
    float* pstop = preds + (size_t)N_NODES * (T_STEPS + 1) * IN_F;  // (768, 4, 2)

    const size_t MAT = (size_t)N_NODES * HID;           // 98304 floats
    float* ws   = (float*)d_ws;
    float* H    = ws;                                   // hidden / h_new
    float* Z    = ws + MAT;
    float* Aa   = ws + 2 * MAT;                         // z @ Wm_src
    float* Bp   = ws + 3 * MAT;                         // z @ Wm_dst + bm
    float* U    = ws + 4 * MAT;
    float* MEAN = ws + 5 * MAT;                         // 128 floats

    setup_kernel<<<384, 256, 0, stream>>>(states, H, preds, pstop);

    dim3 blk(32);
    dim3 g128(N_NODES / 16, HID / 16);                  // (48, 8)
    dim3 g64(N_NODES / 16, IN_F / 16);                  // (48, 4)

    for (int t = 0; t < T_STEPS; ++t) {
        const float* Xst  = (t == 0) ? states : (preds + (size_t)t * IN_F);
        int          ldst = (t == 0) ? IN_F : (T_STEPS + 1) * IN_F;

        // z = [state | hidden] @ We + be
        gemm16_wmma<<<g128, blk, 0, stream>>>(Xst, ldst, IN_F, H, HID, HID,
                                              We, HID, be, Z, HID);
        // a = z @ Wm_src ; b' = z @ Wm_dst + bm
        gemm16_wmma<<<g128, blk, 0, stream>>>(Z, HID, HID, nullptr, 0, 0,
                                              Wm, HID, nullptr, Aa, HID);
        gemm16_wmma<<<g128, blk, 0, stream>>>(Z, HID, HID, nullptr, 0, 0,
                                              Wm + (size_t)HID * HID, HID, bm, Bp, HID);
        // u = b' + max_i( a[i] + edges[i,j]@Wm_e )   (fused TDM + WMMA K=16)
        msgmax_kernel<<<N_NODES / 16, 256, 0, stream>>>(
            edges, Wm + (size_t)2 * HID * HID, Aa, Bp, U);
        // h_new = [z | u] @ Wu + bu   (overwrites H; old H already consumed)
        gemm16_wmma<<<g128, blk, 0, stream>>>(Z, HID, HID, U, HID, HID,
                                              Wu, HID, bu, H, HID);
        // mean + stop head
        mean_kernel<<<1, HID, 0, stream>>>(H, MEAN);
        stop_kernel<<<3, 256, 0, stream>>>(H, MEAN, Wt, bt,
                                           pstop + (size_t)(t + 1) * 2);
        // new_state = [h_new | z] @ Wd + bd  -> written directly into preds[:, t+1, :]
        gemm16_wmma<<<g64, blk, 0, stream>>>(H, HID, HID, Z, HID, HID,
                                             Wd, IN_F, bd,
                                             preds + (size_t)(t + 1) * IN_F,
                                             (T_STEPS + 1) * IN_F);
    }
}